// MixtureDensity_48404281425964
// MI455X (gfx1250) — compile-verified
//
#include <hip/hip_runtime.h>
#include <math.h>

typedef float v2f __attribute__((ext_vector_type(2)));
typedef float v8f __attribute__((ext_vector_type(8)));
typedef int   v2i __attribute__((ext_vector_type(2)));
typedef int   v4i __attribute__((ext_vector_type(4)));

#define NPTS  65536
#define KC    32
#define D     64
#define LDSTR 68   // padded row stride (floats): 16B-aligned rows, conflict-free b64 reads

#if defined(__HIP_DEVICE_COMPILE__) && !__has_builtin(__builtin_amdgcn_wmma_f32_16x16x4_f32)
#error "no __builtin_amdgcn_wmma_f32_16x16x4_f32 on this toolchain (device pass)"
#endif

// CDNA5 async global->LDS copy path (ASYNCcnt-tracked), if the toolchain exposes it.
#if defined(__HIP_DEVICE_COMPILE__) && \
    __has_builtin(__builtin_amdgcn_global_load_async_to_lds_b128) && \
    __has_builtin(__builtin_amdgcn_global_load_async_to_lds_b64)
#define USE_ASYNC 1
#else
#define USE_ASYNC 0
#endif

#if USE_ASYNC
// builtin params: (v4i AS1* src, v4i AS3* dst, Ii offset, Ii cpol) per clang diagnostics
typedef v4i __attribute__((address_space(1))) *g128_t;
typedef v4i __attribute__((address_space(3))) *l128_t;
typedef v2i __attribute__((address_space(1))) *g64_t;
typedef v2i __attribute__((address_space(3))) *l64_t;
#if __has_builtin(__builtin_amdgcn_s_wait_asynccnt)
#define ASYNC_WAIT0() __builtin_amdgcn_s_wait_asynccnt(0)
#else
#define ASYNC_WAIT0() asm volatile("s_wait_asynccnt 0x0" ::: "memory")
#endif
#endif

// ---------------------------------------------------------------------------
// Kernel 1: per-component prep (32 blocks x 64 threads).
// Build L_k (strict lower + exp(diag)+1e-3), invert by per-column forward
// substitution, write Linv_k densely, c_k = Linv_k * mu_k, and
// const_k = -0.5*D*log(2pi) - sum(log(diag)).
// ---------------------------------------------------------------------------
__global__ void md_prep_kernel(const float* __restrict__ mu,
                               const float* __restrict__ ls,
                               float* __restrict__ linv_out,   // [K][64][64]
                               float* __restrict__ c_out,      // [K][64]
                               float* __restrict__ k_out)      // [K]
{
    __shared__ float Ls[D * LDSTR];
    __shared__ float Ms[D * LDSTR];
    __shared__ float red[D];

    const int k = blockIdx.x;
    const int t = threadIdx.x;                 // 0..63
    const float* lsk = ls + (size_t)k * D * D;

    for (int j = 0; j < D; ++j) {
        float v = lsk[t * D + j];
        Ls[t * LDSTR + j] = (j < t) ? v : 0.0f;
    }
    float dg = __expf(lsk[t * D + t]) + 1e-3f;
    Ls[t * LDSTR + t] = dg;
    red[t] = __logf(dg);
    __syncthreads();

    // thread t computes column t of Minv = L^{-1} (touches only its own column)
    for (int i = 0; i < D; ++i) {
        if (i < t) { Ms[i * LDSTR + t] = 0.0f; continue; }
        float s = (i == t) ? 1.0f : 0.0f;
        for (int m = t; m < i; ++m)
            s -= Ls[i * LDSTR + m] * Ms[m * LDSTR + t];
        Ms[i * LDSTR + t] = s / Ls[i * LDSTR + i];
    }
    __syncthreads();

    const float* muk = mu + k * D;
    float cs = 0.0f;
    for (int j = 0; j < D; ++j) {
        float v = Ms[t * LDSTR + j];
        linv_out[(size_t)k * D * D + t * D + j] = v;
        cs += v * muk[j];
    }
    c_out[k * D + t] = cs;

    if (t == 0) {
        float logdet = 0.0f;
        for (int i = 0; i < D; ++i) logdet += red[i];
        k_out[k] = -0.5f * (float)D * 1.8378770664093453f /* log(2*pi) */ - logdet;
    }
}

// ---------------------------------------------------------------------------
// Kernel 2: main pass. 512 blocks x 256 threads (8 waves). Each wave owns a
// 16-point tile; per component k:  Y = Linv_k (64x64) @ X^T (64x16) via
// V_WMMA_F32_16X16X4_F32 (4 row-chunks x 16 K-steps), then maha + online LSE.
// Linv_k is double-buffered in LDS; component k+1 streams in via the CDNA5
// async global->LDS engine (ASYNCcnt) while k's WMMAs execute.
// ---------------------------------------------------------------------------
__global__ void md_logp_kernel(const float* __restrict__ x,
                               const float* __restrict__ linv,
                               const float* __restrict__ cvec,
                               const float* __restrict__ kconst,
                               float* __restrict__ out)
{
    __shared__ float As[2][D * LDSTR];   // Linv_k double buffer, padded rows
    __shared__ float cs[2][D];           // c_k double buffer

    const int t    = threadIdx.x;     // 0..255
    const int lane = t & 31;
    const int wv   = t >> 5;          // wave in block: 0..7
    const int half = lane >> 4;       // 0: lanes 0-15, 1: lanes 16-31
    const int lr   = lane & 15;
    const int tile = blockIdx.x * 8 + wv;
    const int p    = tile * 16 + lr;  // point index (column of B/D)

    // B fragments (x tile) held in registers for all 32 components.
    // 32-bit operand layout: lane half 0 holds K=0,1 ; half 1 holds K=2,3.
    const float2* x2 = (const float2*)x;
    float2 bfrag[16];
#pragma unroll
    for (int s = 0; s < 16; ++s)
        bfrag[s] = x2[(size_t)p * 32 + 2 * s + half];

#if USE_ASYNC
    // -------- async staging: ASYNCcnt-tracked global->LDS, no VGPR traffic --
    auto stage = [&](int k, int b) {
        const float* src = linv + (size_t)k * D * D;
#pragma unroll
        for (int it = 0; it < 4; ++it) {
            int e   = t + it * 256;   // 1024 x b128 chunks
            int row = e >> 4;
            int cq  = e & 15;
            __builtin_amdgcn_global_load_async_to_lds_b128(
                (g128_t)(void*)(src + row * D + cq * 4),
                (l128_t)(void*)&As[b][row * LDSTR + cq * 4], 0, 0);
        }
        if (t < 32)
            __builtin_amdgcn_global_load_async_to_lds_b64(
                (g64_t)(void*)(cvec + k * D + 2 * t),
                (l64_t)(void*)&cs[b][2 * t], 0, 0);
    };
#else
    // -------- fallback: prefetch into VGPRs, commit to LDS later ------------
    float2 pre[8];
    float2 cpre;
    auto stage = [&](int k) {   // issue global loads (latency hidden by compute)
        const float2* g2 = (const float2*)(linv + (size_t)k * D * D);
#pragma unroll
        for (int it = 0; it < 8; ++it)
            pre[it] = g2[t + it * 256];
        if (t < 32)
            cpre = *(const float2*)&cvec[k * D + 2 * t];
    };
    auto commit = [&](int b) {  // VGPR -> LDS
#pragma unroll
        for (int it = 0; it < 8; ++it) {
            int e   = t + it * 256;
            int row = e >> 5;
            int cp  = e & 31;
            *(float2*)&As[b][row * LDSTR + 2 * cp] = pre[it];
        }
        if (t < 32)
            *(float2*)&cs[b][2 * t] = cpre;
    };
#endif

    float mrun = -INFINITY, ssum = 0.0f;

    // ---- prologue: buffer 0 <- component 0 ----
#if USE_ASYNC
    stage(0, 0);
    ASYNC_WAIT0();
    __syncthreads();
#else
    stage(0);
    commit(0);
    __syncthreads();
#endif

    for (int k = 0; k < KC; ++k) {
        const int cur = k & 1;

#if USE_ASYNC
        if (k < KC - 1) stage(k + 1, 1 - cur);   // overlaps with WMMAs below
#else
        if (k < KC - 1) stage(k + 1);            // loads in flight during compute
#endif

        v8f acc[4] = {};
#pragma unroll
        for (int r = 0; r < 4; ++r) {
#pragma unroll
            for (int s = 0; s < 16; ++s) {
                // A fragment: Linv_k[16r + lr][4s + 2*half + {0,1}]
                float2 ap = *(const float2*)&As[cur][(16 * r + lr) * LDSTR + 4 * s + 2 * half];
                v2f a; a[0] = ap.x; a[1] = ap.y;
                v2f b; b[0] = bfrag[s].x; b[1] = bfrag[s].y;
                acc[r] = __builtin_amdgcn_wmma_f32_16x16x4_f32(
                    false, a, false, b, (short)0, acc[r], false, false);
            }
        }

        // Mahalanobis: D layout -> VGPR j holds row M = 16r + 8*half + j, col = lr.
        float part = 0.0f;
#pragma unroll
        for (int r = 0; r < 4; ++r) {
#pragma unroll
            for (int j = 0; j < 8; ++j) {
                float z = acc[r][j] - cs[cur][16 * r + 8 * half + j];
                part += z * z;
            }
        }
        float maha = part + __shfl_xor(part, 16, 32);
        float logp = -0.5f * maha + kconst[k];

        // online log-sum-exp over components
        if (logp > mrun) {
            ssum = ssum * __expf(mrun - logp) + 1.0f;
            mrun = logp;
        } else {
            ssum += __expf(logp - mrun);
        }

#if USE_ASYNC
        if (k < KC - 1) ASYNC_WAIT0();   // next buffer fully in LDS
        __syncthreads();
#else
        __syncthreads();                 // everyone done reading buffers
        if (k < KC - 1) {
            commit(1 - cur);             // s_wait_loadcnt folded in here
            __syncthreads();
        }
#endif
    }

    if (half == 0)
        out[p] = mrun + __logf(ssum);
}

// ---------------------------------------------------------------------------
extern "C" void kernel_launch(void* const* d_in, const int* in_sizes, int n_in,
                              void* d_out, int out_size, void* d_ws, size_t ws_size,
                              hipStream_t stream) {
    const float* x   = (const float*)d_in[0];   // (65536, 64)
    // d_in[1] = log_pi : softmax over size-1 axis == 1.0, unused
    const float* mu  = (const float*)d_in[2];   // (32, 64)
    const float* ls  = (const float*)d_in[3];   // (32, 64, 64)
    float* out = (float*)d_out;                 // (65536,)

    float* w      = (float*)d_ws;
    float* linv   = w;                          // 32*64*64 floats = 512 KB
    float* cvec   = w + KC * D * D;             // 32*64 floats
    float* kconst = cvec + KC * D;              // 32 floats

    md_prep_kernel<<<KC, D, 0, stream>>>(mu, ls, linv, cvec, kconst);
    md_logp_kernel<<<NPTS / (16 * 8), 256, 0, stream>>>(x, linv, cvec, kconst, out);
}